// GAT_41180146434810
// MI455X (gfx1250) — compile-verified
//
#include <hip/hip_runtime.h>
#include <math.h>

typedef __attribute__((ext_vector_type(2))) float v2f;
typedef __attribute__((ext_vector_type(8))) float v8f;

#define THREADS 256

// ---------------------------------------------------------------------------
// helpers
// ---------------------------------------------------------------------------
__device__ __forceinline__ void atomicMaxF(float* addr, float val) {
  // IEEE-754 bit-pattern ordering trick: non-negative -> signed int max,
  // negative -> unsigned int min. Monotone across mixed signs.
  if (val >= 0.0f) {
    atomicMax(reinterpret_cast<int*>(addr), __float_as_int(val));
  } else {
    atomicMin(reinterpret_cast<unsigned int*>(addr), __float_as_uint(val));
  }
}

__global__ void fill_kernel(float* __restrict__ p, long long n, float v) {
  long long i = (long long)blockIdx.x * blockDim.x + threadIdx.x;
  if (i < n) p[i] = v;
}

// ---------------------------------------------------------------------------
// conv1 feature transform: h1[n, c] = sum_k x[n,k] * W1[k,c]   (K = 3)
// ---------------------------------------------------------------------------
__global__ void xw1_kernel(int n_nodes, const float* __restrict__ x,
                           const float* __restrict__ W, float* __restrict__ h) {
  long long i = (long long)blockIdx.x * blockDim.x + threadIdx.x;
  if (i >= (long long)n_nodes * 256) return;
  int n = (int)(i >> 8), c = (int)(i & 255);
  const float* xr = x + (long long)n * 3;
  h[i] = xr[0] * W[c] + xr[1] * W[256 + c] + xr[2] * W[512 + c];
}

// ---------------------------------------------------------------------------
// per-node attention coefficients: asrc[n,h] = <h[n,h,:], a_src[h,:]>
// ---------------------------------------------------------------------------
__global__ void alpha_kernel(int n_nodes, int H, int C,
                             const float* __restrict__ h,
                             const float* __restrict__ a_src,
                             const float* __restrict__ a_dst,
                             float* __restrict__ asrc, float* __restrict__ adst) {
  long long i = (long long)blockIdx.x * blockDim.x + threadIdx.x;
  if (i >= (long long)n_nodes * H) return;
  int n = (int)(i / H), hh = (int)(i % H);
  const float* hr = h + (long long)n * H * C + (long long)hh * C;
  const float* s = a_src + hh * C;
  const float* d = a_dst + hh * C;
  float ss = 0.f, dd = 0.f;
  for (int c = 0; c < C; ++c) {
    float v = hr[c];
    ss += v * s[c];
    dd += v * d[c];
  }
  asrc[i] = ss;
  adst[i] = dd;
}

// ---------------------------------------------------------------------------
// edge pass 1: logits + segment max (self-loops appended at e >= E)
// ---------------------------------------------------------------------------
__global__ void edge_logit_max(const int* __restrict__ ei, int E, int EP, int H,
                               const float* __restrict__ asrc,
                               const float* __restrict__ adst,
                               float* __restrict__ we, float* __restrict__ m) {
  long long i = (long long)blockIdx.x * blockDim.x + threadIdx.x;
  if (i >= (long long)EP * H) return;
  int e = (int)(i / H), h = (int)(i % H);
  int s = (e < E) ? ei[e] : (e - E);
  int d = (e < E) ? ei[E + e] : (e - E);
  float v = asrc[s * H + h] + adst[d * H + h];
  v = (v > 0.f) ? v : 0.2f * v;            // leaky_relu(0.2)
  we[i] = v;
  atomicMaxF(&m[d * H + h], v);
}

// ---------------------------------------------------------------------------
// edge pass 2: exp(logit - max), accumulate z
// ---------------------------------------------------------------------------
__global__ void edge_expsum(const int* __restrict__ ei, int E, int EP, int H,
                            float* __restrict__ we, const float* __restrict__ m,
                            float* __restrict__ z) {
  long long i = (long long)blockIdx.x * blockDim.x + threadIdx.x;
  if (i >= (long long)EP * H) return;
  int e = (int)(i / H), h = (int)(i % H);
  int d = (e < E) ? ei[E + e] : (e - E);
  float w = __expf(we[i] - m[d * H + h]);
  we[i] = w;
  atomicAdd(&z[d * H + h], w);
}

// ---------------------------------------------------------------------------
// edge pass 3: scatter-add of weighted source features (float4 per thread)
// ---------------------------------------------------------------------------
__global__ void edge_aggr(const int* __restrict__ ei, int E, int EP, int H, int C,
                          const float* __restrict__ hfeat,
                          const float* __restrict__ we, float* __restrict__ out) {
  long long i = (long long)blockIdx.x * blockDim.x + threadIdx.x;
  int HC = H * C;
  int chunks = HC >> 2;
  if (i >= (long long)EP * chunks) return;
  int e = (int)(i / chunks);
  int q = (int)(i % chunks);
  int c4 = q << 2;
  int h = c4 / C;
  int s = (e < E) ? ei[e] : (e - E);
  int d = (e < E) ? ei[E + e] : (e - E);
  float w = we[(long long)e * H + h];
  const float4 hv = *reinterpret_cast<const float4*>(hfeat + (long long)s * HC + c4);
  float* o = out + (long long)d * HC + c4;
  atomicAdd(o + 0, hv.x * w);
  atomicAdd(o + 1, hv.y * w);
  atomicAdd(o + 2, hv.z * w);
  atomicAdd(o + 3, hv.w * w);
}

// ---------------------------------------------------------------------------
// normalize by z, add bias, optional ELU
// ---------------------------------------------------------------------------
__global__ void gat_finish(int n_nodes, int H, int C, const float* __restrict__ z,
                           const float* __restrict__ bias, float* __restrict__ out,
                           int do_elu) {
  long long i = (long long)blockIdx.x * blockDim.x + threadIdx.x;
  int HC = H * C;
  if (i >= (long long)n_nodes * HC) return;
  int n = (int)(i / HC), c = (int)(i % HC);
  int h = c / C;
  float v = out[i] / (z[n * H + h] + 1e-16f) + bias[c];
  if (do_elu) v = (v > 0.f) ? v : (__expf(v) - 1.f);
  out[i] = v;
}

// ---------------------------------------------------------------------------
// fp32 WMMA GEMM: C[M,Ncols] = A[M,K] @ B[K,Ncols] (+bias, optional relu)
// One wave computes a 16x64 macro tile -> four independent v_wmma accumulator
// chains per K-step (hides XDL latency, amortizes A loads / address math).
// Requires M%16==0, Ncols%64==0, K%4==0.
// A operand layout (V_WMMA_F32_16X16X4_F32, ISA 7.12.2): lanes 0-15 carry
// K=k,k+1; lanes 16-31 carry K=k+2,k+3; M = lane%16. B mirrored. C/D:
// M = v + 8*(lane/16), N = lane%16.
// ---------------------------------------------------------------------------
__global__ void gemm_wmma_f32(const float* __restrict__ A, const float* __restrict__ B,
                              float* __restrict__ C, int M, int Ncols, int K,
                              const float* __restrict__ bias, int relu) {
  int tilesN4 = Ncols >> 6;                    // 64-wide macro tiles
  int totalTiles = (M >> 4) * tilesN4;
  int wave = blockIdx.x * (blockDim.x >> 5) + (threadIdx.x >> 5);
  if (wave >= totalTiles) return;              // wave-uniform exit: EXEC all-1s
  int tm = wave / tilesN4, tn = wave % tilesN4;
  int lane = threadIdx.x & 31;
  int half = lane >> 4, li = lane & 15;
  const float* Arow = A + (long long)((tm << 4) + li) * K;
  int col0 = (tn << 6) + li;
  v8f acc0 = {}, acc1 = {}, acc2 = {}, acc3 = {};
  for (int k = 0; k < K; k += 4) {
    int ka = k + (half << 1);
    v2f a = *reinterpret_cast<const v2f*>(Arow + ka);
    const float* Bk0 = B + (long long)ka * Ncols + col0;
    const float* Bk1 = Bk0 + Ncols;
    v2f b0; b0.x = Bk0[0];  b0.y = Bk1[0];
    v2f b1; b1.x = Bk0[16]; b1.y = Bk1[16];
    v2f b2; b2.x = Bk0[32]; b2.y = Bk1[32];
    v2f b3; b3.x = Bk0[48]; b3.y = Bk1[48];
    acc0 = __builtin_amdgcn_wmma_f32_16x16x4_f32(false, a, false, b0, (short)0,
                                                 acc0, false, false);
    acc1 = __builtin_amdgcn_wmma_f32_16x16x4_f32(false, a, false, b1, (short)0,
                                                 acc1, false, false);
    acc2 = __builtin_amdgcn_wmma_f32_16x16x4_f32(false, a, false, b2, (short)0,
                                                 acc2, false, false);
    acc3 = __builtin_amdgcn_wmma_f32_16x16x4_f32(false, a, false, b3, (short)0,
                                                 acc3, false, false);
  }
  float bias0 = 0.f, bias1 = 0.f, bias2 = 0.f, bias3 = 0.f;
  if (bias) {
    bias0 = bias[col0];
    bias1 = bias[col0 + 16];
    bias2 = bias[col0 + 32];
    bias3 = bias[col0 + 48];
  }
  for (int v = 0; v < 8; ++v) {
    int r = (tm << 4) + v + (half << 3);
    float v0 = acc0[v] + bias0;
    float v1 = acc1[v] + bias1;
    float v2 = acc2[v] + bias2;
    float v3 = acc3[v] + bias3;
    if (relu) {
      v0 = fmaxf(v0, 0.f); v1 = fmaxf(v1, 0.f);
      v2 = fmaxf(v2, 0.f); v3 = fmaxf(v3, 0.f);
    }
    float* Cr = C + (long long)r * Ncols + col0;
    Cr[0]  = v0;
    Cr[16] = v1;
    Cr[32] = v2;
    Cr[48] = v3;
  }
}

// ---------------------------------------------------------------------------
// graph pooling: pooled[b, 0:256]=max x1, pooled[b, 256:1024]=max x2
// ---------------------------------------------------------------------------
__global__ void pool_max(int n_nodes, const int* __restrict__ batch,
                         const float* __restrict__ x1, const float* __restrict__ x2,
                         float* __restrict__ pooled) {
  long long i = (long long)blockIdx.x * blockDim.x + threadIdx.x;
  if (i >= (long long)n_nodes * 1024) return;
  int n = (int)(i >> 10), j = (int)(i & 1023);
  float v = (j < 256) ? x1[(long long)n * 256 + j]
                      : x2[(long long)n * 768 + (j - 256)];
  atomicMaxF(&pooled[(long long)batch[n] * 1024 + j], v);
}

// ---------------------------------------------------------------------------
// inference BatchNorm1d in-place
// ---------------------------------------------------------------------------
__global__ void bn_kernel(float* __restrict__ xio, const float* __restrict__ g,
                          const float* __restrict__ be, const float* __restrict__ rm,
                          const float* __restrict__ rv, long long total, int F) {
  long long i = (long long)blockIdx.x * blockDim.x + threadIdx.x;
  if (i >= total) return;
  int f = (int)(i % F);
  xio[i] = (xio[i] - rm[f]) * rsqrtf(rv[f] + 1e-5f) * g[f] + be[f];
}

// ---------------------------------------------------------------------------
// final tiny GEMM: out[b, c] = t2[b,:] @ lw3[:, c] + lb3[c]  (NCLS = 2)
// ---------------------------------------------------------------------------
__global__ void final_kernel(int Bg, const float* __restrict__ t2,
                             const float* __restrict__ lw3,
                             const float* __restrict__ lb3, float* __restrict__ out) {
  int i = blockIdx.x * blockDim.x + threadIdx.x;
  if (i >= Bg * 2) return;
  int b = i >> 1, c = i & 1;
  const float* tr = t2 + (long long)b * 256;
  float s = lb3[c];
  for (int k = 0; k < 256; ++k) s += tr[k] * lw3[k * 2 + c];
  out[i] = s;
}

// ---------------------------------------------------------------------------
// host launcher
// ---------------------------------------------------------------------------
extern "C" void kernel_launch(void* const* d_in, const int* in_sizes, int n_in,
                              void* d_out, int out_size, void* d_ws, size_t ws_size,
                              hipStream_t stream) {
  const float* x      = (const float*)d_in[0];
  const int*   ei     = (const int*)d_in[1];
  const int*   batch  = (const int*)d_in[2];
  const float* W1     = (const float*)d_in[3];
  const float* a_src1 = (const float*)d_in[4];
  const float* a_dst1 = (const float*)d_in[5];
  const float* b1     = (const float*)d_in[6];
  const float* W2     = (const float*)d_in[7];
  const float* a_src2 = (const float*)d_in[8];
  const float* a_dst2 = (const float*)d_in[9];
  const float* b2     = (const float*)d_in[10];
  const float* lw1    = (const float*)d_in[11];
  const float* lb1    = (const float*)d_in[12];
  const float* g1     = (const float*)d_in[13];
  const float* be1    = (const float*)d_in[14];
  const float* rm1    = (const float*)d_in[15];
  const float* rv1    = (const float*)d_in[16];
  const float* lw2    = (const float*)d_in[17];
  const float* lb2    = (const float*)d_in[18];
  const float* g2     = (const float*)d_in[19];
  const float* be2    = (const float*)d_in[20];
  const float* rm2    = (const float*)d_in[21];
  const float* rv2    = (const float*)d_in[22];
  const float* lw3    = (const float*)d_in[23];
  const float* lb3    = (const float*)d_in[24];
  float* out = (float*)d_out;
  (void)n_in; (void)ws_size;

  const int N  = in_sizes[0] / 3;    // 50000
  const int E  = in_sizes[1] / 2;    // 400000
  const int EP = E + N;              // edges + self-loops
  const int Bg = out_size / 2;       // 256

  float* ws = (float*)d_ws;
  size_t off = 0;
  auto take = [&](size_t n) { float* p = ws + off; off += n; return p; };
  float* h1  = take((size_t)N * 256);
  float* x1  = take((size_t)N * 256);
  float* h2  = take((size_t)N * 768);
  float* x2  = take((size_t)N * 768);
  float* as1 = take((size_t)N * 4);
  float* ad1 = take((size_t)N * 4);
  float* m1  = take((size_t)N * 4);
  float* z1  = take((size_t)N * 4);
  float* as2 = take((size_t)N * 3);
  float* ad2 = take((size_t)N * 3);
  float* m2  = take((size_t)N * 3);
  float* z2  = take((size_t)N * 3);
  float* we1 = take((size_t)EP * 4);
  float* we2 = take((size_t)EP * 3);
  float* pooled = take((size_t)Bg * 1024);
  float* t1  = take((size_t)Bg * 512);
  float* t2  = take((size_t)Bg * 256);

  auto nb = [](long long n) { return (unsigned)((n + THREADS - 1) / THREADS); };
  const float NEG_INF = -INFINITY;

  // ---- conv1 ----
  xw1_kernel<<<nb((long long)N * 256), THREADS, 0, stream>>>(N, x, W1, h1);
  alpha_kernel<<<nb((long long)N * 4), THREADS, 0, stream>>>(N, 4, 64, h1, a_src1,
                                                             a_dst1, as1, ad1);
  fill_kernel<<<nb((long long)N * 4), THREADS, 0, stream>>>(m1, (long long)N * 4, NEG_INF);
  fill_kernel<<<nb((long long)N * 4), THREADS, 0, stream>>>(z1, (long long)N * 4, 0.f);
  fill_kernel<<<nb((long long)N * 256), THREADS, 0, stream>>>(x1, (long long)N * 256, 0.f);
  edge_logit_max<<<nb((long long)EP * 4), THREADS, 0, stream>>>(ei, E, EP, 4, as1, ad1,
                                                                we1, m1);
  edge_expsum<<<nb((long long)EP * 4), THREADS, 0, stream>>>(ei, E, EP, 4, we1, m1, z1);
  edge_aggr<<<nb((long long)EP * 64), THREADS, 0, stream>>>(ei, E, EP, 4, 64, h1, we1, x1);
  gat_finish<<<nb((long long)N * 256), THREADS, 0, stream>>>(N, 4, 64, z1, b1, x1, 1);

  // ---- conv2 feature GEMM via WMMA: h2 = x1 @ W2 ----
  {
    long long waves = (long long)(N >> 4) * (768 >> 6);
    gemm_wmma_f32<<<nb(waves * 32), THREADS, 0, stream>>>(x1, W2, h2, N, 768, 256,
                                                          nullptr, 0);
  }
  alpha_kernel<<<nb((long long)N * 3), THREADS, 0, stream>>>(N, 3, 256, h2, a_src2,
                                                             a_dst2, as2, ad2);
  fill_kernel<<<nb((long long)N * 3), THREADS, 0, stream>>>(m2, (long long)N * 3, NEG_INF);
  fill_kernel<<<nb((long long)N * 3), THREADS, 0, stream>>>(z2, (long long)N * 3, 0.f);
  fill_kernel<<<nb((long long)N * 768), THREADS, 0, stream>>>(x2, (long long)N * 768, 0.f);
  edge_logit_max<<<nb((long long)EP * 3), THREADS, 0, stream>>>(ei, E, EP, 3, as2, ad2,
                                                                we2, m2);
  edge_expsum<<<nb((long long)EP * 3), THREADS, 0, stream>>>(ei, E, EP, 3, we2, m2, z2);
  edge_aggr<<<nb((long long)EP * 192), THREADS, 0, stream>>>(ei, E, EP, 3, 256, h2, we2, x2);
  gat_finish<<<nb((long long)N * 768), THREADS, 0, stream>>>(N, 3, 256, z2, b2, x2, 0);

  // ---- pooling ----
  fill_kernel<<<nb((long long)Bg * 1024), THREADS, 0, stream>>>(pooled,
                                                               (long long)Bg * 1024,
                                                               NEG_INF);
  pool_max<<<nb((long long)N * 1024), THREADS, 0, stream>>>(N, batch, x1, x2, pooled);

  // ---- MLP head (WMMA GEMMs + BN) ----
  {
    long long waves = (long long)(Bg >> 4) * (512 >> 6);
    gemm_wmma_f32<<<nb(waves * 32), THREADS, 0, stream>>>(pooled, lw1, t1, Bg, 512, 1024,
                                                          lb1, 1);
  }
  bn_kernel<<<nb((long long)Bg * 512), THREADS, 0, stream>>>(t1, g1, be1, rm1, rv1,
                                                             (long long)Bg * 512, 512);
  {
    long long waves = (long long)(Bg >> 4) * (256 >> 6);
    gemm_wmma_f32<<<nb(waves * 32), THREADS, 0, stream>>>(t1, lw2, t2, Bg, 256, 512,
                                                          lb2, 1);
  }
  bn_kernel<<<nb((long long)Bg * 256), THREADS, 0, stream>>>(t2, g2, be2, rm2, rv2,
                                                             (long long)Bg * 256, 256);
  final_kernel<<<nb((long long)Bg * 2), THREADS, 0, stream>>>(Bg, t2, lw3, lb3, out);
}